// GCNEncoder_10694468567653
// MI455X (gfx1250) — compile-verified
//
#include <hip/hip_runtime.h>
#include <hip/hip_bf16.h>

typedef float v2f __attribute__((ext_vector_type(2)));
typedef float v8f __attribute__((ext_vector_type(8)));

#define DFEAT 128

// ---------------------------------------------------------------------------
// Degree / normalization kernels
// ---------------------------------------------------------------------------
__global__ __launch_bounds__(256) void k_deg_init(float* __restrict__ deg, int n) {
    int i = blockIdx.x * blockDim.x + threadIdx.x;
    if (i < n) deg[i] = 1.0f;  // self-loop contributes 1 to every node's degree
}

__global__ __launch_bounds__(256) void k_deg_accum(const long long* __restrict__ dst,
                                                   float* __restrict__ deg, long long e_cnt) {
    long long e = (long long)blockIdx.x * blockDim.x + threadIdx.x;
    if (e < e_cnt) atomicAdd(&deg[dst[e]], 1.0f);
}

__global__ __launch_bounds__(256) void k_rsqrt(float* __restrict__ deg, int n) {
    int i = blockIdx.x * blockDim.x + threadIdx.x;
    if (i < n) deg[i] = rsqrtf(deg[i]);  // deg >= 1 always (self loops)
}

// ---------------------------------------------------------------------------
// GEMM: H[N x 128] = X[N x 128] @ W[128 x 128] using V_WMMA_F32_16X16X4_F32.
// One wave (32 lanes) per 16x16 output tile; K-loop of 32 WMMA ops.
//
// f32 operand layouts (ISA 7.12.2):
//   A 16x4 : lanes 0-15 -> M=0..15, VGPR0/1 = K={0,1}; lanes 16-31 = K={2,3}
//   B 4x16 : lanes 0-15 -> N=0..15, VGPR0/1 = K={0,1}; lanes 16-31 = K={2,3}
//   C 16x16: VGPR r, lanes 0-15 -> (M=r,  N=lane); lanes 16-31 -> (M=r+8, N=lane-16)
// EXEC must be all 1s: grid covers exact multiples of 16 rows (100000 % 16 == 0).
// ---------------------------------------------------------------------------
__global__ __launch_bounds__(32) void k_gemm_wmma(const float* __restrict__ X,
                                                  const float* __restrict__ W,
                                                  float* __restrict__ H) {
    const int tileM = blockIdx.x * 16;
    const int tileN = blockIdx.y * 16;
    const int lane  = threadIdx.x;
    const int half  = lane >> 4;   // 0: K pair {0,1}, 1: K pair {2,3}
    const int l15   = lane & 15;

    const float* arow = X + (size_t)(tileM + l15) * DFEAT + half * 2; // A: row per lane
    const float* bcol = W + (size_t)(half * 2) * DFEAT + tileN + l15; // B: col per lane

    v8f acc = {};
#pragma unroll
    for (int k = 0; k < DFEAT; k += 4) {
        v2f a, b;
        a.x = arow[k];
        a.y = arow[k + 1];
        b.x = bcol[(size_t)k * DFEAT];
        b.y = bcol[(size_t)(k + 1) * DFEAT];
        // (neg_a, A, neg_b, B, c_mod, C, reuse_a, reuse_b)
        acc = __builtin_amdgcn_wmma_f32_16x16x4_f32(false, a, false, b,
                                                    (short)0, acc, false, false);
    }

    float* out = H + (size_t)(tileM + half * 8) * DFEAT + tileN + l15;
#pragma unroll
    for (int r = 0; r < 8; ++r) out[(size_t)r * DFEAT] = acc[r];
}

// ---------------------------------------------------------------------------
// AGG[i,:] = H[i,:] * dinv[i]^2   (self-loop message; doubles as the zero-init)
// One float4 per thread: idx over N*32 float4 slots.
// ---------------------------------------------------------------------------
__global__ __launch_bounds__(256) void k_selfloop(const float* __restrict__ H,
                                                  const float* __restrict__ dinv,
                                                  float* __restrict__ AGG, int n) {
    int idx = blockIdx.x * blockDim.x + threadIdx.x;   // [0, n*32)
    if (idx >= n * 32) return;
    int i    = idx >> 5;
    int lane = idx & 31;
    float w = dinv[i];
    w = w * w;
    float4 v = ((const float4*)(H + (size_t)i * DFEAT))[lane];
    float4 o = make_float4(v.x * w, v.y * w, v.z * w, v.w * w);
    ((float4*)(AGG + (size_t)i * DFEAT))[lane] = o;
}

// ---------------------------------------------------------------------------
// Edge scatter: one wave per edge. Each lane gathers float4 of h[src],
// scales by dinv[src]*dinv[dst], and does 4 global f32 atomic adds to agg[dst].
// ---------------------------------------------------------------------------
__global__ __launch_bounds__(256) void k_scatter(const float* __restrict__ H,
                                                 const float* __restrict__ dinv,
                                                 const long long* __restrict__ src,
                                                 const long long* __restrict__ dst,
                                                 float* __restrict__ AGG, long long e_cnt) {
    long long e = (long long)blockIdx.x * (blockDim.x >> 5) + (threadIdx.x >> 5);
    if (e >= e_cnt) return;
    int lane = threadIdx.x & 31;
    long long s = src[e];
    long long d = dst[e];
    float norm = dinv[s] * dinv[d];
    float4 v = ((const float4*)(H + (size_t)s * DFEAT))[lane];
    float* out = AGG + (size_t)d * DFEAT + (size_t)lane * 4;
    atomicAdd(out + 0, v.x * norm);
    atomicAdd(out + 1, v.y * norm);
    atomicAdd(out + 2, v.z * norm);
    atomicAdd(out + 3, v.w * norm);
}

// ---------------------------------------------------------------------------
// OUT[i,f] = act(AGG[i,f] + bias[f]);  relu for layer 1, identity for layer 2
// ---------------------------------------------------------------------------
__global__ __launch_bounds__(256) void k_bias_act(const float* __restrict__ AGG,
                                                  const float* __restrict__ bias,
                                                  float* __restrict__ OUT,
                                                  int total, int do_relu) {
    int idx = blockIdx.x * blockDim.x + threadIdx.x;
    if (idx >= total) return;
    float v = AGG[idx] + bias[idx & (DFEAT - 1)];
    OUT[idx] = do_relu ? fmaxf(v, 0.0f) : v;
}

// ---------------------------------------------------------------------------
// Host-side launch sequence
// ---------------------------------------------------------------------------
extern "C" void kernel_launch(void* const* d_in, const int* in_sizes, int n_in,
                              void* d_out, int out_size, void* d_ws, size_t ws_size,
                              hipStream_t stream) {
    const float*     x   = (const float*)d_in[0];
    const long long* ei  = (const long long*)d_in[1];
    const float*     W1  = (const float*)d_in[2];
    const float*     b1  = (const float*)d_in[3];
    const float*     W2  = (const float*)d_in[4];
    const float*     b2  = (const float*)d_in[5];

    const int       N = in_sizes[0] / DFEAT;       // 100000
    const long long E = (long long)in_sizes[1] / 2; // 1600000
    const long long* src = ei;
    const long long* dst = ei + E;

    // Workspace layout (floats): dinv[N] | h[N*128] | agg[N*128]
    float* dinv = (float*)d_ws;
    float* h    = dinv + (((size_t)N + 255) & ~(size_t)255);
    float* agg  = h + (size_t)N * DFEAT;
    float* out  = (float*)d_out;

    const int total   = N * DFEAT;
    const int gN      = (N + 255) / 256;
    const int gE      = (int)((E + 255) / 256);
    const int gEw     = (int)((E + 7) / 8);     // 8 edges per 256-thread block
    const int gTot    = (total + 255) / 256;
    const int gSelf   = (N * 32 + 255) / 256;
    dim3 gGemm(N / 16, DFEAT / 16);             // 6250 x 8 tiles, wave32 each

    // Symmetric normalization coefficients (shared by both layers)
    k_deg_init <<<gN,   256, 0, stream>>>(dinv, N);
    k_deg_accum<<<gE,   256, 0, stream>>>(dst, dinv, E);
    k_rsqrt    <<<gN,   256, 0, stream>>>(dinv, N);

    // ---- Layer 1: h = x @ W1 ; agg = scatter(norm * h) ; agg = relu(agg + b1)
    k_gemm_wmma<<<gGemm, 32, 0, stream>>>(x, W1, h);
    k_selfloop <<<gSelf, 256, 0, stream>>>(h, dinv, agg, N);
    k_scatter  <<<gEw,   256, 0, stream>>>(h, dinv, src, dst, agg, E);
    k_bias_act <<<gTot,  256, 0, stream>>>(agg, b1, agg, total, 1);

    // ---- Layer 2: h = agg @ W2 ; out = scatter(norm * h) ; out = out + b2
    k_gemm_wmma<<<gGemm, 32, 0, stream>>>(agg, W2, h);
    k_selfloop <<<gSelf, 256, 0, stream>>>(h, dinv, out, N);
    k_scatter  <<<gEw,   256, 0, stream>>>(h, dinv, src, dst, out, E);
    k_bias_act <<<gTot,  256, 0, stream>>>(out, b2, out, total, 0);
}